// RLQuantumCircuitLayer_49916109914684
// MI455X (gfx1250) — compile-verified
//
#include <hip/hip_runtime.h>
#include <math.h>

// ---------------------------------------------------------------------------
// RLQuantumCircuitLayer for MI455X (gfx1250)
//   B=16384 batch rows, Q=64 qubits, L=16 layers.
//   Roofline: 8 MB total HBM traffic (~0.35 us @ 23.3 TB/s) vs ~1.2 GFLOP of
//   dependent FMA work -> VALU/latency bound. No matmul-shaped ops (4x4
//   block-diagonal rotations, sequential entangle chain) -> WMMA not
//   applicable; the CDNA5 paths that pay are async global<->LDS DMA,
//   LDS-resident state, scalar-cache coefficient loads, split wait counters.
// ---------------------------------------------------------------------------

#define QN  64
#define LN  16
#define TPB 64   // 2 wave32 per block

// ---- vector/address-space types matching the async builtin signatures ------
typedef int v4i __attribute__((__vector_size__(16)));
typedef __attribute__((address_space(1))) v4i gv4i;   // global (HIP __device__)
typedef __attribute__((address_space(3))) v4i lv4i;   // LDS

// ---- CDNA5 async LDS<->global data movement --------------------------------
__device__ __forceinline__ void async_g2l_b128(const void* g, void* l) {
#if __has_builtin(__builtin_amdgcn_global_load_async_to_lds_b128)
  __builtin_amdgcn_global_load_async_to_lds_b128((gv4i*)g, (lv4i*)l, 0, 0);
#else
  unsigned int laddr = (unsigned int)(unsigned long long)l;
  asm volatile("global_load_async_to_lds_b128 %0, %1, off"
               :: "v"(laddr), "v"(g) : "memory");
#endif
}

__device__ __forceinline__ void async_l2g_b128(void* g, void* l) {
#if __has_builtin(__builtin_amdgcn_global_store_async_from_lds_b128)
  __builtin_amdgcn_global_store_async_from_lds_b128((gv4i*)g, (lv4i*)l, 0, 0);
#else
  unsigned int laddr = (unsigned int)(unsigned long long)l;
  asm volatile("global_store_async_from_lds_b128 %0, %1, off"
               :: "v"(g), "v"(laddr) : "memory");
#endif
}

__device__ __forceinline__ void wait_async0() {
#if __has_builtin(__builtin_amdgcn_s_wait_asynccnt)
  __builtin_amdgcn_s_wait_asynccnt(0);
#else
  asm volatile("s_wait_asynccnt 0x0" ::: "memory");
#endif
}

// ---------------------------------------------------------------------------
// Kernel 1: coefficients (full-precision libm; runs once, 1024 threads).
//   coef[l][q][8] = { cos(rz/2), sin(rz/2), cos(ry/2), sin(ry/2),
//                     cos(rx/2), sin(rx/2), sig, 1-sig }
//   sig = sigmoid(ent[l][q-1]) for q>0, 0 for q==0 (qubit 0 never entangles).
// ---------------------------------------------------------------------------
__global__ __launch_bounds__(TPB) void qc_coef_kernel(
    const float* __restrict__ rot,   // (L, Q, 3)  [rx, ry, rz]
    const float* __restrict__ ent,   // (L, Q-1)
    float* __restrict__ coef)        // (L, Q, 8)
{
  int idx = blockIdx.x * blockDim.x + threadIdx.x;   // 0 .. L*Q-1
  if (idx >= LN * QN) return;
  int l = idx >> 6;
  int q = idx & 63;

  float rx = rot[idx * 3 + 0];
  float ry = rot[idx * 3 + 1];
  float rz = rot[idx * 3 + 2];

  float sig = 0.0f;
  if (q > 0) {
    float e = ent[l * (QN - 1) + (q - 1)];
    sig = 1.0f / (1.0f + expf(-e));
  }

  float* c = coef + idx * 8;
  c[0] = cosf(rz * 0.5f);  c[1] = sinf(rz * 0.5f);
  c[2] = cosf(ry * 0.5f);  c[3] = sinf(ry * 0.5f);
  c[4] = cosf(rx * 0.5f);  c[5] = sinf(rx * 0.5f);
  c[6] = sig;              c[7] = 1.0f - sig;
}

// ---------------------------------------------------------------------------
// Kernel 2: main simulation.  One thread = one batch row.
//   LDS: state[QN][TPB] float4 (64 KB) + xstage[TPB][QN] float (16 KB) = 80 KB
//   state[q*TPB+tid] accesses put lanes 0..15 across all 64 banks (stride 4
//   banks/lane) -> conflict-free b128.  xstage row accesses are staggered by
//   tid so concurrent lanes hit distinct bank groups.
// ---------------------------------------------------------------------------
struct SmemLayout {
  float4 state[QN * TPB];   // state[q*TPB + tid] = {re0, im0, re1, im1}
  float  xstage[TPB * QN];  // raw x rows on entry, |amp1|^2 rows on exit
};

__global__ __launch_bounds__(TPB) void qc_main_kernel(
    const float* __restrict__ x,     // (B, Q)
    const float* __restrict__ coef,  // (L, Q, 8) -- uniform -> scalar loads
    float* __restrict__ out)         // (B, Q)
{
  extern __shared__ char smem_raw[];
  SmemLayout& sm = *(SmemLayout*)smem_raw;

  const int tid  = threadIdx.x;        // 0..63
  const int lane = tid & 31;
  const int wave = tid >> 5;           // 0..1
  const long long rowBase = (long long)blockIdx.x * TPB;

  // ---- async-stage this wave's 32 rows of x (8 KB contiguous) into LDS ----
  {
    const char* g = (const char*)(x + (rowBase + (long long)wave * 32) * QN);
    char*       l = (char*)(sm.xstage + (wave * 32) * QN);
#pragma unroll
    for (int i = 0; i < 16; ++i) {
      int off = (i * 32 + lane) * 16;
      async_g2l_b128(g + off, l + off);
    }
  }
  wait_async0();   // per-wave; each thread only reads rows staged by its wave

  // ---- init state: amp0 = cos(pi/2*x), amp1 = sin(pi/2*x)  (hw sin/cos) ---
  {
    const float4* xrow = (const float4*)(sm.xstage + tid * QN);
#pragma unroll
    for (int g4 = 0; g4 < QN / 4; ++g4) {
      int qg = (g4 + tid) & 15;            // stagger: spread lanes over banks
      float4 xv = xrow[qg];
#pragma unroll
      for (int j = 0; j < 4; ++j) {
        float xe = (j == 0) ? xv.x : (j == 1) ? xv.y : (j == 2) ? xv.z : xv.w;
        float h  = xe * 1.57079632679489662f;   // in [0, pi/2): hw-safe range
        float4 s;
        s.x = __cosf(h); s.y = 0.0f;   // amp0
        s.z = __sinf(h); s.w = 0.0f;   // amp1
        sm.state[(qg * 4 + j) * TPB + tid] = s;
      }
    }
  }

  // ---- 16 layers: Rz,Ry,Rx rotation + sequential entangle chain -----------
  for (int l = 0; l < LN; ++l) {
    const float* cl = coef + l * (QN * 8);   // uniform address -> s_load
    float ctrl = 0.0f;                       // |amp1| carry along the chain
#pragma unroll 4
    for (int q = 0; q < QN; ++q) {
      float4 s = sm.state[q * TPB + tid];
      float pzr = cl[q * 8 + 0], pzi = cl[q * 8 + 1];
      float cy  = cl[q * 8 + 2], sy  = cl[q * 8 + 3];
      float cx  = cl[q * 8 + 4], sx  = cl[q * 8 + 5];
      float sig = cl[q * 8 + 6], w0  = cl[q * 8 + 7];   // w0 = 1 - sig

      // Rz: a0 = s0*pz, a1 = s1*conj(pz)
      float a0r = s.x * pzr - s.y * pzi;
      float a0i = s.x * pzi + s.y * pzr;
      float a1r = s.z * pzr + s.w * pzi;
      float a1i = s.w * pzr - s.z * pzi;
      // Ry (real coefficients)
      float b0r = cy * a0r - sy * a1r;
      float b0i = cy * a0i - sy * a1i;
      float b1r = sy * a0r + cy * a1r;
      float b1i = sy * a0i + cy * a1i;
      // Rx (sx multiplies by i)
      float c0r = cx * b0r + sx * b1i;
      float c0i = cx * b0i - sx * b1r;
      float c1r = sx * b0i + cx * b1r;
      float c1i = cx * b1i - sx * b0r;
      // entangle: new = (1-sig)*c + sig*ctrl*swap(c)   (sig==0 for q==0)
      float wc  = sig * ctrl;
      float n0r = w0 * c0r + wc * c1r;
      float n0i = w0 * c0i + wc * c1i;
      float n1r = w0 * c1r + wc * c0r;
      float n1i = w0 * c1i + wc * c0i;
      ctrl = sqrtf(n1r * n1r + n1i * n1i);

      float4 ns; ns.x = n0r; ns.y = n0i; ns.z = n1r; ns.w = n1i;
      sm.state[q * TPB + tid] = ns;
    }
  }

  // ---- output: |amp1|^2, staged (staggered, vectorized) then async-stored --
  {
    float4* orow = (float4*)(sm.xstage + tid * QN);
#pragma unroll
    for (int g4 = 0; g4 < QN / 4; ++g4) {
      int qg = (g4 + tid) & 15;            // stagger: spread lanes over banks
      float4 p;
      float4 s0 = sm.state[(qg * 4 + 0) * TPB + tid];
      float4 s1 = sm.state[(qg * 4 + 1) * TPB + tid];
      float4 s2 = sm.state[(qg * 4 + 2) * TPB + tid];
      float4 s3 = sm.state[(qg * 4 + 3) * TPB + tid];
      p.x = s0.z * s0.z + s0.w * s0.w;
      p.y = s1.z * s1.z + s1.w * s1.w;
      p.z = s2.z * s2.z + s2.w * s2.w;
      p.w = s3.z * s3.z + s3.w * s3.w;
      orow[qg] = p;
    }
  }
  __syncthreads();   // flush DS stores before the async engine reads LDS
  {
    char* g = (char*)(out + (rowBase + (long long)wave * 32) * QN);
    char* l = (char*)(sm.xstage + (wave * 32) * QN);
#pragma unroll
    for (int i = 0; i < 16; ++i) {
      int off = (i * 32 + lane) * 16;
      async_l2g_b128(g + off, l + off);
    }
  }
  wait_async0();
}

// ---------------------------------------------------------------------------
// Host launcher
// ---------------------------------------------------------------------------
extern "C" void kernel_launch(void* const* d_in, const int* in_sizes, int n_in,
                              void* d_out, int out_size, void* d_ws, size_t ws_size,
                              hipStream_t stream) {
  (void)n_in; (void)out_size; (void)ws_size;
  const float* x   = (const float*)d_in[0];   // (B, Q) f32
  const float* rot = (const float*)d_in[1];   // (L, Q, 3) f32
  const float* ent = (const float*)d_in[2];   // (L, Q-1) f32
  float* out  = (float*)d_out;
  float* coef = (float*)d_ws;                 // needs L*Q*8*4 = 32 KB scratch

  const int B = in_sizes[0] / QN;             // 16384

  qc_coef_kernel<<<(LN * QN + TPB - 1) / TPB, TPB, 0, stream>>>(rot, ent, coef);
  qc_main_kernel<<<B / TPB, TPB, sizeof(SmemLayout), stream>>>(x, coef, out);
}